// TrajNet_38783554682945
// MI455X (gfx1250) — compile-verified
//
#include <hip/hip_runtime.h>
#include <hip/hip_bf16.h>

typedef float v2f __attribute__((ext_vector_type(2)));
typedef float v8f __attribute__((ext_vector_type(8)));

namespace {

constexpr float kRhoObs = 1.2f;
constexpr float kRhoEq  = 10.0f;
constexpr float kWSm    = 10.0f;
constexpr int   kIters  = 300;
constexpr float kSumX   = 130.79f;  // sum of X_OBS column values
constexpr float kSumY   = 69.2f;    // sum of Y_OBS column values

enum Slab {
  LX = 0, LY, EQX, EQY, LBX, LBY, DCA, DSA,
  SOLX, SOLY, LINX, LINY, TXS, TYS, RESX, RESY, NSLAB
};

__device__ __forceinline__ v8f wmma4(v2f a, v2f b, v8f c) {
  // D = A(16x4,f32) * B(4x16,f32) + C(16x16,f32)  -- full-precision CDNA5 WMMA
  return __builtin_amdgcn_wmma_f32_16x16x4_f32(false, a, false, b, (short)0, c,
                                               false, false);
}

// C(16x16) += A(16x12 in LDS, row-major stride 16) * B(12x16 in B-layout regs)
// A-layout: lane L holds row (L&15); VGPR j of k-step t holds K = 4t + j + 2*(L>>4)
__device__ __forceinline__ v8f gemmK12(const float* A, const v2f* B, v8f c,
                                       int arow, int hi) {
#pragma unroll
  for (int t = 0; t < 3; ++t) {
    v2f a;
    a[0] = A[arow * 16 + 4 * t + 2 * hi + 0];
    a[1] = A[arow * 16 + 4 * t + 2 * hi + 1];
    c = wmma4(a, B[t], c);
  }
  return c;
}

__global__ __launch_bounds__(64) void traj_admm_kernel(
    const float* __restrict__ xg, const float* __restrict__ bg,
    const float* __restrict__ W1, const float* __restrict__ b1,
    const float* __restrict__ W2, const float* __restrict__ b2,
    const float* __restrict__ Pg, const float* __restrict__ Pd,
    const float* __restrict__ Pdd, float* __restrict__ out) {
  __shared__ float slab[2][NSLAB][256];  // two waves, private 16x16 f32 slabs
  __shared__ float cAeq[66];             // 6x11
  __shared__ float cGeq[121];            // 11x11  A_eq^T A_eq
  __shared__ float cInv[121];            // 11x11  inv(-cost)
  __shared__ float cColSum[11];          // column sums of P
  __shared__ float gj[11 * 22];          // Gauss-Jordan workspace

  const int tid  = threadIdx.x;
  const int wv   = tid >> 5;
  const int lane = tid & 31;
  const int col  = lane & 15;  // C/D & B layout: N column for this lane
  const int hi   = lane >> 4;  // half-wave selector

  // ---- zero all per-wave state (pads matter for padded GEMMs) ----------
  for (int i = tid; i < 2 * NSLAB * 256; i += 64) (&slab[0][0][0])[i] = 0.0f;

  // ---- block-local constant prep (11x11 inverse etc.; tiny, serial) ----
  if (tid == 0) {
    for (int n = 0; n < 11; ++n) {
      cAeq[0 * 11 + n] = Pg[0 * 11 + n];
      cAeq[1 * 11 + n] = Pd[0 * 11 + n];
      cAeq[2 * 11 + n] = Pdd[0 * 11 + n];
      cAeq[3 * 11 + n] = Pg[9 * 11 + n];
      cAeq[4 * 11 + n] = Pd[9 * 11 + n];
      cAeq[5 * 11 + n] = Pdd[9 * 11 + n];
    }
    for (int i = 0; i < 11; ++i)
      for (int j = 0; j < 11; ++j) {
        float gdd = 0.f, gp = 0.f, ge = 0.f;
        for (int k = 0; k < 10; ++k) {
          gdd += Pdd[k * 11 + i] * Pdd[k * 11 + j];
          gp  += Pg[k * 11 + i] * Pg[k * 11 + j];
        }
        for (int r = 0; r < 6; ++r) ge += cAeq[r * 11 + i] * cAeq[r * 11 + j];
        cGeq[i * 11 + j] = ge;
        // A_obs^T A_obs == 4 * P^T P  (A_obs = tile(P,4))
        float cost = kWSm * gdd + kRhoObs * 4.0f * gp + kRhoEq * ge;
        gj[i * 22 + j]      = -cost;
        gj[i * 22 + 11 + j] = (i == j) ? 1.0f : 0.0f;
      }
    for (int cpiv = 0; cpiv < 11; ++cpiv) {  // Gauss-Jordan, partial pivot
      int piv = cpiv;
      float best = fabsf(gj[cpiv * 22 + cpiv]);
      for (int r = cpiv + 1; r < 11; ++r) {
        float v = fabsf(gj[r * 22 + cpiv]);
        if (v > best) { best = v; piv = r; }
      }
      if (piv != cpiv)
        for (int j = 0; j < 22; ++j) {
          float t = gj[cpiv * 22 + j];
          gj[cpiv * 22 + j] = gj[piv * 22 + j];
          gj[piv * 22 + j] = t;
        }
      float iv = 1.0f / gj[cpiv * 22 + cpiv];
      for (int j = 0; j < 22; ++j) gj[cpiv * 22 + j] *= iv;
      for (int r = 0; r < 11; ++r)
        if (r != cpiv) {
          float f = gj[r * 22 + cpiv];
          for (int j = 0; j < 22; ++j) gj[r * 22 + j] -= f * gj[cpiv * 22 + j];
        }
    }
    for (int i = 0; i < 11; ++i)
      for (int j = 0; j < 11; ++j) cInv[i * 11 + j] = gj[i * 22 + 11 + j];
    for (int n = 0; n < 11; ++n) {
      float s = 0.f;
      for (int k = 0; k < 10; ++k) s += Pg[k * 11 + n];
      cColSum[n] = s;
    }
  }
  __syncthreads();

  // ---- per-batch-row setup: MLP -> b_gen -> rho_eq*(b@A_eq), linbase ----
  if (tid < 32) {
    const int lw = tid >> 4;  // wave slab
    const int lr = tid & 15;  // local row
    const int row = blockIdx.x * 32 + tid;
    float bp[12];
    for (int n = 0; n < 12; ++n) bp[n] = b2[n];
    for (int j = 0; j < 64; ++j) {
      float acc = b1[j];
      for (int k = 0; k < 16; ++k) acc += xg[row * 16 + k] * W1[j * 16 + k];
      float h = fmaxf(acc, 0.0f);
      for (int n = 0; n < 12; ++n) bp[n] += h * W2[n * 64 + j];
    }
    float bgen[12];
    for (int n = 0; n < 12; ++n)
      bgen[n] = (n == 3 || n == 9) ? bp[n] : bg[row * 12 + n];  // MASK blend
    float (*SB)[256] = slab[lw];
    for (int n = 0; n < 11; ++n) {
      float ex = 0.f, ey = 0.f;
      for (int r = 0; r < 6; ++r) {
        ex += bgen[r]     * cAeq[r * 11 + n];
        ey += bgen[6 + r] * cAeq[r * 11 + n];
      }
      ex *= kRhoEq; ey *= kRhoEq;
      SB[EQX][lr * 16 + n] = ex;
      SB[EQY][lr * 16 + n] = ey;
      SB[LBX][lr * 16 + n] = -ex - kRhoObs * kSumX * cColSum[n];
      SB[LBY][lr * 16 + n] = -ey - kRhoObs * kSumY * cColSum[n];
    }
    // init carry: d=1, alpha=0  =>  sum_o d*cos = 4, sum_o d*sin = 0
    for (int k = 0; k < 10; ++k) SB[DCA][lr * 16 + k] = 4.0f;
  }

  // ---- loop-invariant B operands in B-layout registers (zero padded) ----
  // B-layout: VGPR j of k-step t, lane L: K = 4t + j + 2*(L>>4), N = L&15
  v2f bP[3], bInv[3], bPT[3], bGq[3];
#pragma unroll
  for (int t = 0; t < 3; ++t)
#pragma unroll
    for (int j = 0; j < 2; ++j) {
      int kk = 4 * t + j + 2 * hi;
      bP[t][j]   = (kk < 10 && col < 11) ? -kRhoObs * Pg[kk * 11 + col] : 0.0f;
      bInv[t][j] = (kk < 11 && col < 11) ? cInv[kk * 11 + col]          : 0.0f;
      bPT[t][j]  = (kk < 11 && col < 10) ? Pg[col * 11 + kk]            : 0.0f;
      bGq[t][j]  = (kk < 11 && col < 11) ? -kRhoEq * cGeq[kk * 11 + col] : 0.0f;
    }
  __syncthreads();

  float (*S)[256] = slab[wv];
  const int ar = col;  // A-layout row for this lane
  const float xv[4] = {-10.0f, 100.79f, 30.0f, 10.0f};
  const float yv[4] = {-10.0f, 100.0f, -30.8f, 10.0f};

  for (int it = 0; it < kIters; ++it) {
    // ---- lin = (linbase - lx) + dcasum @ (-rho_obs * P) ----
    v8f cx, cy;
#pragma unroll
    for (int r = 0; r < 8; ++r) {
      int idx = (r + 8 * hi) * 16 + col;
      cx[r] = S[LBX][idx] - S[LX][idx];
      cy[r] = S[LBY][idx] - S[LY][idx];
    }
    cx = gemmK12(S[DCA], bP, cx, ar, hi);
    cy = gemmK12(S[DSA], bP, cy, ar, hi);
#pragma unroll
    for (int r = 0; r < 8; ++r) {
      int idx = (r + 8 * hi) * 16 + col;
      S[LINX][idx] = cx[r];
      S[LINY][idx] = cy[r];
    }
    __syncthreads();

    // ---- sol = lin @ inv(neg_cost) ----
#pragma unroll
    for (int r = 0; r < 8; ++r) { cx[r] = 0.0f; cy[r] = 0.0f; }
    cx = gemmK12(S[LINX], bInv, cx, ar, hi);
    cy = gemmK12(S[LINY], bInv, cy, ar, hi);
#pragma unroll
    for (int r = 0; r < 8; ++r) {
      int idx = (r + 8 * hi) * 16 + col;
      S[SOLX][idx] = cx[r];
      S[SOLY][idx] = cy[r];
    }
    __syncthreads();

    // ---- t = sol @ P^T ----
#pragma unroll
    for (int r = 0; r < 8; ++r) { cx[r] = 0.0f; cy[r] = 0.0f; }
    cx = gemmK12(S[SOLX], bPT, cx, ar, hi);
    cy = gemmK12(S[SOLY], bPT, cy, ar, hi);
#pragma unroll
    for (int r = 0; r < 8; ++r) {
      int idx = (r + 8 * hi) * 16 + col;
      S[TXS][idx] = cx[r];
      S[TYS][idx] = cy[r];
    }
    __syncthreads();

    // ---- elementwise: circle projection (A_OBS_R = B_OBS_R = 1) ----
    // d*cos(atan2(ws,wc)) = wc        if ||w|| >= 1   (d = max(||w||,1))
    //                     = wc/||w||  otherwise
#pragma unroll
    for (int q = 0; q < 5; ++q) {
      int p  = lane + 32 * q;  // 160 (row,k) pairs
      int rr = p / 10, k = p % 10;
      float tx = S[TXS][rr * 16 + k];
      float ty = S[TYS][rr * 16 + k];
      float ndca = 0.f, ndsa = 0.f;
#pragma unroll
      for (int o = 0; o < 4; ++o) {
        float wc = tx - xv[o];
        float ws = ty - yv[o];
        float rn = sqrtf(wc * wc + ws * ws);
        float dce, dse;
        if (rn > 1.0f)        { dce = wc; dse = ws; }
        else if (rn > 1e-20f) { float iv = 1.0f / rn; dce = wc * iv; dse = ws * iv; }
        else                  { dce = 1.0f; dse = 0.0f; }
        ndca += dce; ndsa += dse;
      }
      S[DCA][rr * 16 + k]  = ndca;                       // carry to next iter
      S[DSA][rr * 16 + k]  = ndsa;
      S[RESX][rr * 16 + k] = 4.0f * tx - kSumX - ndca;   // sum_o res_x[o,k]
      S[RESY][rr * 16 + k] = 4.0f * ty - kSumY - ndsa;
    }
    __syncthreads();

    // ---- lx' = (lx + eqx) + ressum@(-rho_obs*P) + sol@(-rho_eq*Geq) ----
#pragma unroll
    for (int r = 0; r < 8; ++r) {
      int idx = (r + 8 * hi) * 16 + col;
      cx[r] = S[LX][idx] + S[EQX][idx];
      cy[r] = S[LY][idx] + S[EQY][idx];
    }
    cx = gemmK12(S[RESX], bP, cx, ar, hi);
    cy = gemmK12(S[RESY], bP, cy, ar, hi);
    cx = gemmK12(S[SOLX], bGq, cx, ar, hi);
    cy = gemmK12(S[SOLY], bGq, cy, ar, hi);
#pragma unroll
    for (int r = 0; r < 8; ++r) {
      int idx = (r + 8 * hi) * 16 + col;
      S[LX][idx] = cx[r];
      S[LY][idx] = cy[r];
    }
    __syncthreads();
  }

  // ---- output: x_pred = last tx, y_pred = last ty, concat (512x20) ----
  const int rowbase = blockIdx.x * 32 + wv * 16;
  for (int p = lane; p < 16 * 20; p += 32) {
    int r = p / 20, c = p % 20;
    float v = (c < 10) ? S[TXS][r * 16 + c] : S[TYS][r * 16 + (c - 10)];
    out[(rowbase + r) * 20 + c] = v;
  }
}

}  // namespace

extern "C" void kernel_launch(void* const* d_in, const int* in_sizes, int n_in,
                              void* d_out, int out_size, void* d_ws, size_t ws_size,
                              hipStream_t stream) {
  (void)in_sizes; (void)n_in; (void)out_size; (void)d_ws; (void)ws_size;
  traj_admm_kernel<<<dim3(16), dim3(64), 0, stream>>>(
      (const float*)d_in[0],  // x      (512,16)
      (const float*)d_in[1],  // b      (512,12)
      (const float*)d_in[2],  // W1     (64,16)
      (const float*)d_in[3],  // b1     (64,)
      (const float*)d_in[4],  // W2     (12,64)
      (const float*)d_in[5],  // b2     (12,)
      (const float*)d_in[6],  // P      (10,11)
      (const float*)d_in[7],  // Pdot   (10,11)
      (const float*)d_in[8],  // Pddot  (10,11)
      (float*)d_out);         // out    (512,20)
}